// TrilinearIntepolation_8985071583872
// MI455X (gfx1250) — compile-verified
//
#include <hip/hip_runtime.h>

// Problem constants (from reference setup_inputs)
#define B_   2
#define F_   32
#define D_   32
#define H_   128
#define W_   128
#define HG_  256
#define WG_  256
#define DHW_ (D_ * H_ * W_)          // 524288
#define NPTS (B_ * HG_ * WG_)        // 131072

#if defined(__HIP_DEVICE_COMPILE__) && __has_builtin(__builtin_amdgcn_global_load_async_to_lds_b32)
#define HAVE_ASYNC_LDS 1
#endif

__device__ __forceinline__ void wait_asynccnt0() {
#if defined(__HIP_DEVICE_COMPILE__)
#if __has_builtin(__builtin_amdgcn_s_wait_asynccnt)
  __builtin_amdgcn_s_wait_asynccnt(0);
#else
  asm volatile("s_wait_asynccnt 0" ::: "memory");
#endif
#endif
}

// ---------------------------------------------------------------------------
// Pass 1: transpose [B,F,D,H,W] -> [B,D,H,W,F]  (makes gathers cacheline-granular)
// One thread per voxel (b,z,y,x): 32 coalesced strided reads, 128B contiguous write.
// ---------------------------------------------------------------------------
__global__ void __launch_bounds__(256) trilerp_transpose(
    const float* __restrict__ feats, float* __restrict__ tf) {
  const int idx = blockIdx.x * 256 + threadIdx.x;    // ((b*32+z)*128+y)*128+x
  const int b   = idx >> 19;                         // / DHW_
  const int zyx = idx & (DHW_ - 1);
  const float* src = feats + ((size_t)b << 24) + zyx;  // b*F*DHW = b*2^24
  float v[F_];
#pragma unroll
  for (int f = 0; f < F_; ++f) v[f] = src[(size_t)f * DHW_];
  float4* dst = (float4*)(tf + (size_t)idx * F_);
#pragma unroll
  for (int j = 0; j < F_ / 4; ++j)
    dst[j] = make_float4(v[4 * j], v[4 * j + 1], v[4 * j + 2], v[4 * j + 3]);
}

// ---------------------------------------------------------------------------
// Shared helper: grid coord -> corner weights/indices (replicates reference,
// including the round-half-even quirk: xi==127 / yi==127 are invalid -> 0,
// z is clamped to 31).
// ---------------------------------------------------------------------------
__device__ __forceinline__ void trilerp_setup(float gx, float gy, float gz,
                                              int* x0, int* y0, int* z0,
                                              float wx[2], float wy[2], float wz[2]) {
  const float x = (fminf(fmaxf(gx, -1.f), 1.f) + 1.f) * 0.5f * 127.f;
  const float y = (fminf(fmaxf(gy, -1.f), 1.f) + 1.f) * 0.5f * 127.f;
  const float z = (fminf(fmaxf(gz, -1.f), 1.f) + 1.f) * 0.5f * 31.f;
  const float xf = floorf(x), yf = floorf(y), zf = floorf(z);
  const float u = x - xf, v = y - yf, w = z - zf;
  *x0 = (int)xf; *y0 = (int)yf; *z0 = (int)zf;
  wx[0] = 1.f - u; wx[1] = u;
  wy[0] = 1.f - v; wy[1] = v;
  wz[0] = 1.f - w; wz[1] = w;
}

// ---------------------------------------------------------------------------
// Pass 2: gather from transposed layout. Block = 256 thr = 32 points,
// 8 lanes/point, each lane owns 4 features (one b128 load per corner).
// Grid coords staged via CDNA5 async global->LDS; output transposed through
// LDS for fully coalesced stores.
// ---------------------------------------------------------------------------
__global__ void __launch_bounds__(256) trilerp_gather(
    const float* __restrict__ tf, const float* __restrict__ grid,
    float* __restrict__ out) {
  __shared__ float sGrid[96];          // 32 points * 3 coords
  __shared__ float sOut[32][33];       // padded: conflict-free both phases

  const int t = threadIdx.x;
  const size_t gbase = (size_t)blockIdx.x * 96;

#ifdef HAVE_ASYNC_LDS
  if (t < 96) {   // exactly 3 full waves -> EXEC all-ones for issuing waves
    __builtin_amdgcn_global_load_async_to_lds_b32(
        (__attribute__((address_space(1))) int*)(grid + gbase + t),
        (__attribute__((address_space(3))) int*)(&sGrid[t]),
        0, 0);
  }
  wait_asynccnt0();
#else
  if (t < 96) sGrid[t] = grid[gbase + t];
#endif
  __syncthreads();

  const int pl = t >> 3;                    // point within block (0..31)
  const int r  = t & 7;                     // feature quad (0..7)
  const int point = blockIdx.x * 32 + pl;
  const int b   = point >> 16;
  const float gx = sGrid[pl * 3 + 0];
  const float gy = sGrid[pl * 3 + 1];
  const float gz = sGrid[pl * 3 + 2];

  int x0, y0, z0; float wx[2], wy[2], wz[2];
  trilerp_setup(gx, gy, gz, &x0, &y0, &z0, wx, wy, wz);

  const float* base = tf + ((size_t)b << 24) + (r << 2);   // +r*4 feats
  float4 acc = make_float4(0.f, 0.f, 0.f, 0.f);
#pragma unroll
  for (int dx = 0; dx < 2; ++dx)
#pragma unroll
    for (int dy = 0; dy < 2; ++dy)
#pragma unroll
      for (int dz = 0; dz < 2; ++dz) {
        const int xi = x0 + dx, yi = y0 + dy, zi = z0 + dz;
        const float wc = (xi < 127 && yi < 127) ? wx[dx] * wy[dy] * wz[dz] : 0.f;
        const int xc = xi > 127 ? 127 : xi;
        const int yc = yi > 127 ? 127 : yi;
        const int zc = zi > 31 ? 31 : zi;
        const size_t off = ((size_t)((zc * H_ + yc) * W_ + xc)) << 5;  // *F_
        const float4 val = *(const float4*)(base + off);
        acc.x = fmaf(wc, val.x, acc.x);
        acc.y = fmaf(wc, val.y, acc.y);
        acc.z = fmaf(wc, val.z, acc.z);
        acc.w = fmaf(wc, val.w, acc.w);
      }

  sOut[pl][(r << 2) + 0] = acc.x;
  sOut[pl][(r << 2) + 1] = acc.y;
  sOut[pl][(r << 2) + 2] = acc.z;
  sOut[pl][(r << 2) + 3] = acc.w;
  __syncthreads();

  // Coalesced output: lanes of a wave cover 32 consecutive points, fixed f.
  const int fgrp = t >> 5;                  // 0..7
  const int pw   = t & 31;
  const int pt2  = blockIdx.x * 32 + pw;
  const int b2   = pt2 >> 16;
  const int rem2 = pt2 & 0xFFFF;            // hg*256+wg
#pragma unroll
  for (int k = 0; k < 4; ++k) {
    const int f = fgrp + (k << 3);
    out[((size_t)(b2 * F_ + f) << 16) + rem2] = sOut[pw][f];
  }
}

// ---------------------------------------------------------------------------
// Fallback (if workspace too small): direct gather from native layout.
// ---------------------------------------------------------------------------
__global__ void __launch_bounds__(256) trilerp_direct(
    const float* __restrict__ feats, const float* __restrict__ grid,
    float* __restrict__ out) {
  const int p = blockIdx.x * 256 + threadIdx.x;
  const int b = p >> 16;
  const int rem = p & 0xFFFF;
  const float gx = grid[(size_t)p * 3 + 0];
  const float gy = grid[(size_t)p * 3 + 1];
  const float gz = grid[(size_t)p * 3 + 2];

  int x0, y0, z0; float wx[2], wy[2], wz[2];
  trilerp_setup(gx, gy, gz, &x0, &y0, &z0, wx, wy, wz);

  int   off[8];
  float wc[8];
#pragma unroll
  for (int c = 0; c < 8; ++c) {
    const int dx = c >> 2, dy = (c >> 1) & 1, dz = c & 1;
    const int xi = x0 + dx, yi = y0 + dy, zi = z0 + dz;
    wc[c] = (xi < 127 && yi < 127) ? wx[dx] * wy[dy] * wz[dz] : 0.f;
    const int xc = xi > 127 ? 127 : xi;
    const int yc = yi > 127 ? 127 : yi;
    const int zc = zi > 31 ? 31 : zi;
    off[c] = (zc * H_ + yc) * W_ + xc;
  }

  const float* base = feats + ((size_t)b << 24);
#pragma unroll 4
  for (int f = 0; f < F_; ++f) {
    const float* fb = base + (size_t)f * DHW_;
    float acc = 0.f;
#pragma unroll
    for (int c = 0; c < 8; ++c) acc = fmaf(wc[c], fb[off[c]], acc);
    out[((size_t)(b * F_ + f) << 16) + rem] = acc;
  }
}

extern "C" void kernel_launch(void* const* d_in, const int* in_sizes, int n_in,
                              void* d_out, int out_size, void* d_ws, size_t ws_size,
                              hipStream_t stream) {
  const float* feats = (const float*)d_in[0];   // [2,32,32,128,128] f32
  const float* grid  = (const float*)d_in[1];   // [2,256,256,3] f32
  float* out = (float*)d_out;                   // [2,32,256,256] f32

  const size_t TBYTES = (size_t)B_ * D_ * H_ * W_ * F_ * sizeof(float); // 128 MiB
  if (d_ws != nullptr && ws_size >= TBYTES) {
    float* tf = (float*)d_ws;
    trilerp_transpose<<<(B_ * DHW_) / 256, 256, 0, stream>>>(feats, tf);
    trilerp_gather<<<NPTS / 32, 256, 0, stream>>>(tf, grid, out);
  } else {
    trilerp_direct<<<NPTS / 256, 256, 0, stream>>>(feats, grid, out);
  }
}